// LSM_16887811408660
// MI455X (gfx1250) — compile-verified
//
#include <hip/hip_runtime.h>
#include <math.h>

typedef __attribute__((ext_vector_type(2))) float v2f;
typedef __attribute__((ext_vector_type(8))) float v8f;

#define N_NODES 100000
#define F_DIM   500
#define C_CLS   40
#define H_DIM   256
#define HX_DIM  64
#define NE      1600000
#define NT      320          // H + HX combined output cols of GEMM1
#define BM      32           // node rows per workgroup
#define KC      100          // K chunk staged in LDS (500 = 5 * 100)
#define XPAD    102          // LDS row stride for x chunk (even -> 8B aligned v2f)
#define ZPAD    326          // LDS row stride for z (h|xe) buffer
#define LPAD    48           // LDS row stride for logits (40 padded to 3 tiles)

// ws layout (floats)
#define WS_PB1  0            // 125 ksteps * 20 ntiles * 64 = 160000
#define WS_PB2  160000       // 64 ksteps * 3 ntiles * 64   = 12288
#define WS_SQ   172288       // N
#define WS_SK   272288       // N

__device__ __forceinline__ v8f wmma_f32(v2f a, v2f b, v8f c) {
  // D = A(16x4 f32) * B(4x16 f32) + C(16x16 f32); 8-arg VOP3P form.
  return __builtin_amdgcn_wmma_f32_16x16x4_f32(
      /*neg_a=*/false, a, /*neg_b=*/false, b,
      /*c_mod=*/(short)0, c, /*reuse_a=*/false, /*reuse_b=*/false);
}

// ---------------------------------------------------------------------------
// Repack weights into WMMA B-fragment layout.
// Frag = 32 lanes * float2, lane(0-15): col=n0+lane, K={k,k+1};
//        lane(16-31): col=n0+lane-16, K={k+2,k+3}.  256B contiguous per frag.
// ---------------------------------------------------------------------------
__global__ __launch_bounds__(256)
void repack_kernel(const float* __restrict__ fc1_w,
                   const float* __restrict__ xenc_w,
                   const float* __restrict__ fc2_w,
                   float* __restrict__ pB1,
                   float* __restrict__ pB2) {
  int g = blockIdx.x * blockDim.x + threadIdx.x;
  const int NB1 = 125 * 20 * 32;  // ksteps * ntiles * lanes
  if (g < NB1) {
    int kstep = g / 640;
    int rem   = g % 640;
    int ntile = rem / 32;
    int lane  = rem % 32;
    int col   = ntile * 16 + (lane & 15);
    int k     = kstep * 4 + 2 * (lane >> 4);
    const float* src = (col < H_DIM) ? (fc1_w  + (size_t)col * F_DIM)
                                     : (xenc_w + (size_t)(col - H_DIM) * F_DIM);
    pB1[(size_t)g * 2 + 0] = src[k];
    pB1[(size_t)g * 2 + 1] = src[k + 1];
  } else {
    int h = g - NB1;
    if (h < 64 * 3 * 32) {
      int kstep = h / 96;
      int rem   = h % 96;
      int ntile = rem / 32;
      int lane  = rem % 32;
      int col   = ntile * 16 + (lane & 15);
      int k     = kstep * 4 + 2 * (lane >> 4);
      float v0 = 0.f, v1 = 0.f;
      if (col < C_CLS) {
        v0 = fc2_w[(size_t)col * H_DIM + k];
        v1 = fc2_w[(size_t)col * H_DIM + k + 1];
      }
      pB2[(size_t)h * 2 + 0] = v0;
      pB2[(size_t)h * 2 + 1] = v1;
    }
  }
}

// ---------------------------------------------------------------------------
// Fused: z = relu(x@Wc^T + b)  ->  logits = h@fc2^T + b  ->  log_softmax,
// y_prob mix, s_q/s_k.  One 32-row node block per workgroup, 8 waves.
// ---------------------------------------------------------------------------
__global__ __launch_bounds__(256)
void fused_encoder(const float* __restrict__ x,
                   const int*   __restrict__ ylab,
                   const unsigned char* __restrict__ mask,
                   const float* __restrict__ fc1_b,
                   const float* __restrict__ fc2_b,
                   const float* __restrict__ xenc_b,
                   const float* __restrict__ pexy_w,
                   const float* __restrict__ pB1,
                   const float* __restrict__ pB2,
                   float* __restrict__ sq,
                   float* __restrict__ sk,
                   float* __restrict__ ylp) {
  __shared__ float s_x[BM * XPAD];   // staged x chunk
  __shared__ float s_z[BM * ZPAD];   // relu(h | xe)
  __shared__ float s_l[BM * LPAD];   // logits (padded to 48)

  const int tid  = threadIdx.x;
  const int wave = tid >> 5;
  const int lane = tid & 31;
  const int lrow = lane & 15;
  const int half = lane >> 4;
  const int row0 = blockIdx.x * BM;

  // ----- phase 1: GEMM1, waves = 2(M) x 4(N), each wave 16 rows x 80 cols
  const int mi = wave >> 2;   // 0..1
  const int nj = wave & 3;    // 0..3
  v8f acc[5] = {};

  for (int kc = 0; kc < F_DIM; kc += KC) {
    __syncthreads();
    for (int i = tid; i < BM * KC; i += 256) {
      int r = i / KC, c = i % KC;
      int g = row0 + r;
      if (g >= N_NODES) g = N_NODES - 1;
      s_x[r * XPAD + c] = x[(size_t)g * F_DIM + kc + c];
    }
    __syncthreads();
    const float* arow = &s_x[(mi * 16 + lrow) * XPAD + 2 * half];
    for (int kk = 0; kk < KC; kk += 4) {
      v2f a = *(const v2f*)(arow + kk);
      int kstep = (kc + kk) >> 2;
      const float* bb = pB1 + ((size_t)kstep * 20 + nj * 5) * 64 + lane * 2;
#pragma unroll
      for (int t = 0; t < 5; ++t) {
        v2f b = *(const v2f*)(bb + (size_t)t * 64);
        acc[t] = wmma_f32(a, b, acc[t]);
      }
    }
  }
  // bias + relu -> LDS z buffer
#pragma unroll
  for (int t = 0; t < 5; ++t) {
    int col = nj * 80 + t * 16 + lrow;
    float bias = (col < H_DIM) ? fc1_b[col] : xenc_b[col - H_DIM];
#pragma unroll
    for (int v = 0; v < 8; ++v) {
      int r = mi * 16 + v + 8 * half;
      float val = acc[t][v] + bias;
      s_z[r * ZPAD + col] = val > 0.f ? val : 0.f;
    }
  }
  __syncthreads();

  // ----- phase 2: GEMM2 (h[32x256] @ fc2^T[256x48]) -> 6 tiles on waves 0..5
  if (wave < 6) {
    int tmi = wave / 3, tn = wave % 3;
    v8f c2 = {};
    const float* arow = &s_z[(tmi * 16 + lrow) * ZPAD + 2 * half];
    for (int k = 0; k < H_DIM; k += 4) {
      v2f a = *(const v2f*)(arow + k);
      v2f b = *(const v2f*)(pB2 + ((size_t)(k >> 2) * 3 + tn) * 64 + lane * 2);
      c2 = wmma_f32(a, b, c2);
    }
    int col = tn * 16 + lrow;
    float bias = (col < C_CLS) ? fc2_b[col] : 0.f;
#pragma unroll
    for (int v = 0; v < 8; ++v) {
      int r = tmi * 16 + v + 8 * half;
      s_l[r * LPAD + col] = c2[v] + bias;
    }
  }
  __syncthreads();

  // ----- phase 3: per-row log_softmax, y_prob mix, s_q/s_k (threads 0..31)
  if (tid < BM) {
    int n = row0 + tid;
    if (n < N_NODES) {
      const float* lr = &s_l[tid * LPAD];
      float m = lr[0];
      for (int c = 1; c < C_CLS; ++c) m = fmaxf(m, lr[c]);
      float s = 0.f;
      for (int c = 0; c < C_CLS; ++c) s += __expf(lr[c] - m);
      float lse = m + __logf(s);
      bool  tm  = mask[n] != 0;
      int   yy  = ylab[n];
      const float* wx = pexy_w;                // [xq(64) | xk(64)]
      const float* wy = pexy_w + 2 * HX_DIM;   // [yq(40) | yk(40)]
      const float* xe = &s_z[tid * ZPAD + H_DIM];
      float sqv = 0.f, skv = 0.f;
      for (int j = 0; j < HX_DIM; ++j) {
        float v = xe[j];
        sqv += v * wx[j];
        skv += v * wx[HX_DIM + j];
      }
      for (int c = 0; c < C_CLS; ++c) {
        float lp = lr[c] - lse;
        ylp[(size_t)n * C_CLS + c] = lp;
        float p = tm ? (c == yy ? 1.f : 0.f) : __expf(lp);
        sqv += p * wy[c];
        skv += p * wy[C_CLS + c];
      }
      sq[n] = sqv;
      sk[n] = skv;
    }
  }
}

// ---------------------------------------------------------------------------
// Edge scores: out[e] = s_q[src] + s_k[dst] + b   (pos and neg sets)
// ---------------------------------------------------------------------------
__global__ __launch_bounds__(256)
void edge_kernel(const int* __restrict__ ei,
                 const int* __restrict__ en,
                 const float* __restrict__ sq,
                 const float* __restrict__ sk,
                 const float* __restrict__ pexy_b,
                 float* __restrict__ out) {
  int e = blockIdx.x * blockDim.x + threadIdx.x;
  if (e < NE) {
    float b = pexy_b[0];
    out[e]      = sq[ei[e]] + sk[ei[NE + e]] + b;
    out[NE + e] = sq[en[e]] + sk[en[NE + e]] + b;
  }
}

extern "C" void kernel_launch(void* const* d_in, const int* in_sizes, int n_in,
                              void* d_out, int out_size, void* d_ws, size_t ws_size,
                              hipStream_t stream) {
  (void)in_sizes; (void)n_in; (void)out_size; (void)ws_size;

  const float* x      = (const float*)d_in[0];
  const int*   ylab   = (const int*)d_in[1];
  const unsigned char* mask = (const unsigned char*)d_in[2];  // jax bool = 1 byte
  const int*   ei     = (const int*)d_in[3];
  const int*   en     = (const int*)d_in[4];
  const float* fc1_w  = (const float*)d_in[5];
  const float* fc1_b  = (const float*)d_in[6];
  const float* fc2_w  = (const float*)d_in[7];
  const float* fc2_b  = (const float*)d_in[8];
  const float* xenc_w = (const float*)d_in[9];
  const float* xenc_b = (const float*)d_in[10];
  const float* pexy_w = (const float*)d_in[11];
  const float* pexy_b = (const float*)d_in[12];

  float* ws  = (float*)d_ws;
  float* pB1 = ws + WS_PB1;
  float* pB2 = ws + WS_PB2;
  float* sq  = ws + WS_SQ;
  float* sk  = ws + WS_SK;

  float* out = (float*)d_out;
  float* ylp = out + 2 * (size_t)NE;   // y_log_prob after pos|neg edge scores

  int total_pack = 125 * 20 * 32 + 64 * 3 * 32;
  repack_kernel<<<(total_pack + 255) / 256, 256, 0, stream>>>(fc1_w, xenc_w, fc2_w, pB1, pB2);

  fused_encoder<<<N_NODES / BM, 256, 0, stream>>>(x, ylab, mask, fc1_b, fc2_b, xenc_b,
                                                  pexy_w, pB1, pB2, sq, sk, ylp);

  edge_kernel<<<(NE + 255) / 256, 256, 0, stream>>>(ei, en, sq, sk, pexy_b, out);
}